// TorchVisionDetector_29025388987103
// MI455X (gfx1250) — compile-verified
//
#include <hip/hip_runtime.h>
#include <hip/hip_bf16.h>

#define B_BATCH 16
#define NBOX    2000
#define NPAD    2048
#define NT      125          // 2000 / 16 column tiles (16-bit mask words)
#define IOU_THR 0.7f

typedef __attribute__((ext_vector_type(2))) float v2f;
typedef __attribute__((ext_vector_type(8))) float v8f;

// ---------------------------------------------------------------------------
// Kernel 1: per-batch max_coord reduction + bitonic argsort(-scores) +
// build sorted, class-offset boxes and their areas.
// ---------------------------------------------------------------------------
__global__ __launch_bounds__(256) void nms_prep_sort(
    const float* __restrict__ boxes, const float* __restrict__ scores,
    const int* __restrict__ labels, float* __restrict__ sBoxes,
    float* __restrict__ sArea, int* __restrict__ order) {
  const int b = blockIdx.x;
  const int tid = threadIdx.x;
  __shared__ unsigned long long keys[NPAD];
  __shared__ float red[256];

  const float* bx = boxes + (size_t)b * NBOX * 4;
  const float* sc = scores + (size_t)b * NBOX;
  const int* lb = labels + (size_t)b * NBOX;

  // max over all coords of this batch (jnp.max(boxes))
  float m = -1e30f;
  for (int i = tid; i < NBOX * 4; i += 256) m = fmaxf(m, bx[i]);
  red[tid] = m;
  __syncthreads();
  for (int s = 128; s > 0; s >>= 1) {
    if (tid < s) red[tid] = fmaxf(red[tid], red[tid + s]);
    __syncthreads();
  }
  const float off_scale = red[0] + 1.0f;

  // keys: descending score (positive floats -> monotone uint bits), stable by idx
  for (int i = tid; i < NPAD; i += 256) {
    unsigned long long key;
    if (i < NBOX) {
      unsigned sb = __float_as_uint(sc[i]);
      key = (((unsigned long long)(sb ^ 0xFFFFFFFFu)) << 32) | (unsigned)i;
    } else {
      key = ~0ull;  // padding sorts last
    }
    keys[i] = key;
  }
  __syncthreads();

  // bitonic sort ascending over 2048 u64 keys
  for (unsigned k = 2; k <= NPAD; k <<= 1) {
    for (unsigned j = k >> 1; j > 0; j >>= 1) {
      for (unsigned t = tid; t < NPAD; t += 256) {
        unsigned p = t ^ j;
        if (p > t) {
          unsigned long long x = keys[t], y = keys[p];
          bool asc = ((t & k) == 0);
          if ((x > y) == asc) { keys[t] = y; keys[p] = x; }
        }
      }
      __syncthreads();
    }
  }

  // emit sorted order, offset boxes, areas
  for (int p = tid; p < NBOX; p += 256) {
    unsigned long long key = keys[p];
    int orig = (int)(unsigned)(key & 0xFFFFFFFFu);
    order[(size_t)b * NBOX + p] = orig;
    float offv = (float)lb[orig] * off_scale;
    float x1 = bx[orig * 4 + 0] + offv;
    float y1 = bx[orig * 4 + 1] + offv;
    float x2 = bx[orig * 4 + 2] + offv;
    float y2 = bx[orig * 4 + 3] + offv;
    float* sbp = sBoxes + ((size_t)b * NBOX + p) * 4;
    sbp[0] = x1; sbp[1] = y1; sbp[2] = x2; sbp[3] = y2;
    sArea[(size_t)b * NBOX + p] = (x2 - x1) * (y2 - y1);
  }
}

// ---------------------------------------------------------------------------
// Kernel 2: suppression mask. One wave owns a 16x16 tile. The (area_i+area_j)
// union base term comes from V_WMMA_F32_16X16X4_F32 (rank-2 outer sum); the
// intersection min/max work follows the WMMA C-layout (lane = column,
// VGPR r = row r (+8 for upper half-wave)). 16-bit masks per (row, colTile):
// exclusive ownership -> no atomics.
// ---------------------------------------------------------------------------
__global__ __launch_bounds__(256) void nms_iou_mask(
    const float* __restrict__ sBoxes, const float* __restrict__ sArea,
    unsigned short* __restrict__ supp) {
  const int rt = blockIdx.x;   // row tile 0..124
  const int b = blockIdx.y;    // batch
  const int tid = threadIdx.x;
  __shared__ float rbox[16][4];
  __shared__ float rarea[16];

  const int ri0 = rt * 16;
  const float* batchBoxes = sBoxes + (size_t)b * NBOX * 4;
  const float* batchArea = sArea + (size_t)b * NBOX;

  if (tid < 16) {
    rbox[tid][0] = batchBoxes[(ri0 + tid) * 4 + 0];
    rbox[tid][1] = batchBoxes[(ri0 + tid) * 4 + 1];
    rbox[tid][2] = batchBoxes[(ri0 + tid) * 4 + 2];
    rbox[tid][3] = batchBoxes[(ri0 + tid) * 4 + 3];
    rarea[tid] = batchArea[ri0 + tid];
  }
  __syncthreads();

  const int w = tid >> 5;          // wave id 0..7
  const int lane = tid & 31;
  const int halfSel = lane >> 4;   // 0: rows ri0..+7 ; 1: rows ri0+8..+15
  const int cl = lane & 15;        // column within tile
  const float rowA = rarea[cl];    // A[m][0] for lanes 0..15 (lane==cl there)
  unsigned short* suppB = supp + (size_t)b * NBOX * NT;

  for (int ct = rt + w; ct < NT; ct += 8) {   // upper triangle incl. diagonal
    const int col = ct * 16 + cl;
    const float* cb = batchBoxes + (size_t)col * 4;
    const float cx1 = cb[0], cy1 = cb[1], cx2 = cb[2], cy2 = cb[3];
    const float ca = batchArea[col];

    // A (16x4): rows [area_i, 1, 0, 0].  Documented f32 A layout:
    // lanes 0-15 carry K=0,1 (VGPR0,1); lanes 16-31 carry K=2,3 (zeros).
    v2f av, bv;
    av[0] = halfSel ? 0.0f : rowA;
    av[1] = halfSel ? 0.0f : 1.0f;
    // B (4x16): cols [1, area_j, X, X]. Filled so either K-half striping of
    // B is correct (A's K=2,3 are zero, so upper-K B content is don't-care).
    bv[0] = halfSel ? ca : 1.0f;
    bv[1] = ca;
    v8f cz = {};
    // D[m][n] = area_row[m] + area_col[n]
    v8f ub = __builtin_amdgcn_wmma_f32_16x16x4_f32(
        false, av, false, bv, (short)0, cz, false, false);

#pragma unroll
    for (int r = 0; r < 8; ++r) {
      const int rowLocal = halfSel * 8 + r;
      const int mrow = ri0 + rowLocal;
      const float rx1 = rbox[rowLocal][0], ry1 = rbox[rowLocal][1];
      const float rx2 = rbox[rowLocal][2], ry2 = rbox[rowLocal][3];
      float iw = fminf(rx2, cx2) - fmaxf(rx1, cx1);
      float ih = fminf(ry2, cy2) - fmaxf(ry1, cy1);
      iw = fmaxf(iw, 0.0f);
      ih = fmaxf(ih, 0.0f);
      const float inter = iw * ih;
      const float uni = ub[r] - inter;                 // area_i+area_j-inter
      // iou > THR  <=>  inter > THR * max(union, 1e-9)   (both sides >= 0)
      const bool sup = (inter > IOU_THR * fmaxf(uni, 1e-9f)) && (col > mrow);
      const unsigned msk = __builtin_amdgcn_ballot_w32(sup);
      if (lane == 0)
        suppB[(size_t)(ri0 + r) * NT + ct] = (unsigned short)(msk & 0xFFFFu);
      else if (lane == 1)
        suppB[(size_t)(ri0 + 8 + r) * NT + ct] = (unsigned short)(msk >> 16);
    }
  }
}

// ---------------------------------------------------------------------------
// Kernel 3: sequential greedy scan, one wave (32 threads) per batch.
// Removed-mask lives in LDS (125 words, 16 bits used each). Next suppression
// row is prefetched (global_prefetch_b8). Only words >= i>>4 are ORed, so
// never-written lower-triangle mask words are never read.
// ---------------------------------------------------------------------------
__global__ __launch_bounds__(32) void nms_scan(
    const unsigned short* __restrict__ supp, unsigned* __restrict__ keepSorted) {
  const int b = blockIdx.x;
  const int lane = threadIdx.x;
  __shared__ unsigned R[NT];
  for (int w = lane; w < NT; w += 32) R[w] = 0u;
  __syncthreads();

  const unsigned short* rowBase = supp + (size_t)b * NBOX * NT;
  unsigned* keepB = keepSorted + (size_t)b * NBOX;

  for (int i = 0; i < NBOX; ++i) {
    if ((i + 1 < NBOX) && (lane < 4)) {
      // 250B row -> 4 prefetches cover it at 64B strides
      __builtin_prefetch(rowBase + (size_t)(i + 1) * NT + lane * 32, 0, 0);
    }
    const unsigned removed = (R[i >> 4] >> (i & 15)) & 1u;
    if (removed == 0u) {
      const unsigned short* row = rowBase + (size_t)i * NT;
      for (int w = (i >> 4) + lane; w < NT; w += 32)
        R[w] |= (unsigned)row[w];
      if (lane == 0) keepB[i] = 1u;
    } else {
      if (lane == 0) keepB[i] = 0u;
    }
  }
}

// ---------------------------------------------------------------------------
// Kernel 4: scatter masked outputs via the sort permutation.
// d_out layout (floats): boxes[128000] | labels[32000] | scores[32000] |
//                        idx[32000] | keep[32000]
// ---------------------------------------------------------------------------
__global__ __launch_bounds__(256) void nms_scatter_out(
    const float* __restrict__ boxes, const float* __restrict__ scores,
    const int* __restrict__ labels, const int* __restrict__ order,
    const unsigned* __restrict__ keepSorted, float* __restrict__ out) {
  const int t = blockIdx.x * blockDim.x + threadIdx.x;
  if (t >= B_BATCH * NBOX) return;
  const int b = t / NBOX;
  const int orig = order[t];
  const int o = b * NBOX + orig;
  const unsigned k = keepSorted[t];
  const float kf = k ? 1.0f : 0.0f;

  float* boxes_out = out;
  float* labels_out = out + (size_t)B_BATCH * NBOX * 4;
  float* scores_out = labels_out + (size_t)B_BATCH * NBOX;
  float* idx_out = scores_out + (size_t)B_BATCH * NBOX;
  float* km_out = idx_out + (size_t)B_BATCH * NBOX;

  boxes_out[o * 4 + 0] = boxes[o * 4 + 0] * kf;
  boxes_out[o * 4 + 1] = boxes[o * 4 + 1] * kf;
  boxes_out[o * 4 + 2] = boxes[o * 4 + 2] * kf;
  boxes_out[o * 4 + 3] = boxes[o * 4 + 3] * kf;
  labels_out[o] = k ? (float)labels[o] : -1.0f;
  scores_out[o] = scores[o] * kf;
  idx_out[o] = (float)b;
  km_out[o] = kf;
}

// ---------------------------------------------------------------------------
extern "C" void kernel_launch(void* const* d_in, const int* in_sizes, int n_in,
                              void* d_out, int out_size, void* d_ws, size_t ws_size,
                              hipStream_t stream) {
  const float* boxes = (const float*)d_in[0];   // (16,2000,4) f32
  const float* scores = (const float*)d_in[1];  // (16,2000)   f32
  const int* labels = (const int*)d_in[2];      // (16,2000)   i32
  (void)in_sizes; (void)n_in; (void)out_size; (void)ws_size;

  // workspace layout (bytes)
  unsigned char* ws = (unsigned char*)d_ws;
  float* sBoxes = (float*)(ws);                            // 512000
  float* sArea = (float*)(ws + 512000);                    // 128000
  int* order = (int*)(ws + 640000);                        // 128000
  unsigned* keepSorted = (unsigned*)(ws + 768000);         // 128000
  unsigned short* supp = (unsigned short*)(ws + 896000);   // 8,000,000

  nms_prep_sort<<<B_BATCH, 256, 0, stream>>>(boxes, scores, labels,
                                             sBoxes, sArea, order);
  nms_iou_mask<<<dim3(NT, B_BATCH), 256, 0, stream>>>(sBoxes, sArea, supp);
  nms_scan<<<B_BATCH, 32, 0, stream>>>(supp, keepSorted);
  nms_scatter_out<<<(B_BATCH * NBOX + 255) / 256, 256, 0, stream>>>(
      boxes, scores, labels, order, keepSorted, (float*)d_out);
}